// BoxAttention_42640435315260
// MI455X (gfx1250) — compile-verified
//
#include <hip/hip_runtime.h>
#include <stdint.h>
#include <math.h>

// ---------------- problem constants (match reference) ----------------
#define D_MODEL   256
#define NUM_HEAD  8
#define NUM_LEVEL 4
#define NUM_POINT 4
#define HEAD_DIM  32
#define B_SZ      4
#define L_LEN     21760                 // 128^2+64^2+32^2+16^2, divisible by 16
#define M_ROWS    (B_SZ * L_LEN)        // 87040, divisible by 16

typedef float v2f __attribute__((ext_vector_type(2)));
typedef float v8f __attribute__((ext_vector_type(8)));

// ----------------------------------------------------------------------
// One wave computes a 16x64 fp32 strip of  C = A(MxK) * W^T(NxK) with
// V_WMMA_F32_16X16X4_F32: one A fragment is reused across 4 column tiles
// (5 float2 loads feed 4 WMMAs).  Operand layout (ISA 7.12.2):
//   A 16x4 : lane m=lane&15, VGPR0/1 = K = 2*(lane>>4) + {0,1}
//   B 4x16 : lane n=lane&15, VGPR0/1 = K = 2*(lane>>4) + {0,1}  (B[k][n]=W[n][k])
//   C/D    : VGPR r -> row r + 8*(lane>>4), col lane&15
// ----------------------------------------------------------------------
#define NT 4   // column tiles per wave (strip width = 64)

static __device__ __forceinline__ void wmma_strip_f32(const float* __restrict__ A,
                                                      const float* __restrict__ W,
                                                      int K, int row_base, int col_base,
                                                      int lane, v8f acc[NT])
{
    const int mlo = lane & 15;
    const int sel = lane >> 4;
    const float* arow = A + (size_t)(row_base + mlo) * K + 2 * sel;
    const float* brow = W + (size_t)(col_base + mlo) * K + 2 * sel;

#pragma unroll
    for (int t = 0; t < NT; ++t)
        acc[t] = (v8f){0.f, 0.f, 0.f, 0.f, 0.f, 0.f, 0.f, 0.f};

#pragma unroll 4
    for (int k = 0; k < K; k += 4) {
        const v2f a = *(const v2f*)(arow + k);
#pragma unroll
        for (int t = 0; t < NT; ++t) {
            const v2f b = *(const v2f*)(brow + (size_t)t * 16 * K + k);
            acc[t] = __builtin_amdgcn_wmma_f32_16x16x4_f32(
                /*neg_a=*/false, a, /*neg_b=*/false, b,
                /*c_mod=*/(short)0, acc[t], /*reuse_a=*/false, /*reuse_b=*/false);
        }
    }
}

// ---------------- value projection: scatter to (B, head, L2, 32) -------
__global__ void k_vproj(const float* __restrict__ value,
                        const float* __restrict__ Wv,
                        const float* __restrict__ bv,
                        const uint8_t* __restrict__ vmask,
                        float* __restrict__ vproj)
{
    const int lane = threadIdx.x;
    const int row_base = blockIdx.x * 16;
    const int col_base = blockIdx.y * (16 * NT);
    v8f acc[NT];
    wmma_strip_f32(value, Wv, D_MODEL, row_base, col_base, lane, acc);

    const int sel = lane >> 4;
    const int nlo = lane & 15;
    // all 16 rows of the tile share one batch (L_LEN % 16 == 0)
    const int b      = row_base / L_LEN;             // scalar
    const int l_base = row_base - b * L_LEN;         // scalar

    uint8_t msk[8];
#pragma unroll
    for (int r = 0; r < 8; ++r) msk[r] = vmask[row_base + r + 8 * sel];

#pragma unroll
    for (int t = 0; t < NT; ++t) {
        const int n    = col_base + t * 16 + nlo;
        const int head = n >> 5;
        const int dim  = n & 31;
        const float bias = bv[n];
        float* dst = vproj + ((size_t)(b * NUM_HEAD + head) * L_LEN + l_base) * HEAD_DIM + dim;
#pragma unroll
        for (int r = 0; r < 8; ++r) {
            const int l_off = r + 8 * sel;
            float v = acc[t][r] + bias;
            if (msk[r]) v = 0.f;
            dst[(size_t)l_off * HEAD_DIM] = v;
        }
    }
}

// ---------------- generic GEMM + bias, row-major C(M x N) --------------
__global__ void k_gemm_bias(const float* __restrict__ A,
                            const float* __restrict__ W,
                            const float* __restrict__ bias,
                            float* __restrict__ C, int N)
{
    const int lane = threadIdx.x;
    const int row_base = blockIdx.x * 16;
    const int col_base = blockIdx.y * (16 * NT);
    v8f acc[NT];
    wmma_strip_f32(A, W, D_MODEL, row_base, col_base, lane, acc);

    const int sel = lane >> 4;
    const int nlo = lane & 15;
#pragma unroll
    for (int t = 0; t < NT; ++t) {
        const int n = col_base + t * 16 + nlo;
        const float bvn = bias[n];
#pragma unroll
        for (int r = 0; r < 8; ++r) {
            const int m = row_base + r + 8 * sel;
            C[(size_t)m * N + n] = acc[t][r] + bvn;
        }
    }
}

// ---------------- attn GEMM fused with per-head softmax ----------------
// N = 128; the 64-wide strip covers 4 heads, each 16-wide group is one
// head's full logit set -> softmax entirely within the block via LDS.
__global__ void k_attn(const float* __restrict__ query,
                       const float* __restrict__ Wa,
                       const float* __restrict__ ba,
                       float* __restrict__ attn_out)   // (M, 128) softmaxed
{
    __shared__ float tile[16][16 * NT + 1];
    const int lane = threadIdx.x;
    const int row_base = blockIdx.x * 16;
    const int col_base = blockIdx.y * (16 * NT);
    v8f acc[NT];
    wmma_strip_f32(query, Wa, D_MODEL, row_base, col_base, lane, acc);

    const int sel = lane >> 4;
    const int nlo = lane & 15;
#pragma unroll
    for (int t = 0; t < NT; ++t) {
        const float bvn = ba[col_base + t * 16 + nlo];
#pragma unroll
        for (int r = 0; r < 8; ++r)
            tile[r + 8 * sel][t * 16 + nlo] = acc[t][r] + bvn;
    }
    __syncthreads();

    if (lane < 16) {
        float* dst = attn_out + (size_t)(row_base + lane) * 128 + col_base;
#pragma unroll
        for (int g = 0; g < NT; ++g) {          // one head-group of 16 each
            float mx = -INFINITY;
#pragma unroll
            for (int j = 0; j < 16; ++j) mx = fmaxf(mx, tile[lane][g * 16 + j]);
            float e[16];
            float s = 0.f;
#pragma unroll
            for (int j = 0; j < 16; ++j) { e[j] = expf(tile[lane][g * 16 + j] - mx); s += e[j]; }
            const float inv = 1.f / s;
#pragma unroll
            for (int j = 0; j < 16; ++j) dst[g * 16 + j] = e[j] * inv;
        }
    }
}

// ---------------- box decode + bilinear sampling -----------------------
// one wave per (b, l, head); lane = channel (HEAD_DIM == warpSize == 32)
__global__ void k_sample(const float* __restrict__ vproj,     // (B, head, L2, 32)
                         const float* __restrict__ attn,      // (M, 128) softmaxed
                         const float* __restrict__ off,       // (M, 128) raw
                         const float* __restrict__ refw,      // (M, 4)
                         const float* __restrict__ vratio,    // (B, 4, 2)
                         const float* __restrict__ kidx,      // (4, 2)
                         float* __restrict__ out_pre)         // (M, 256)
{
    const int q    = blockIdx.x;
    const int head = threadIdx.x >> 5;
    const int lane = threadIdx.x & 31;
    const int b    = q / L_LEN;

    const float cx = refw[q * 4 + 0], cy = refw[q * 4 + 1];
    const float sw = refw[q * 4 + 2], sh = refw[q * 4 + 3];
    const float* ap = attn + (size_t)q * 128 + head * 16;
    const float* op = off  + (size_t)q * 128 + head * 16;

    const int HW[NUM_LEVEL] = {128, 64, 32, 16};
    const int ST[NUM_LEVEL] = {0, 16384, 20480, 21504};

    float acc = 0.f;
#pragma unroll
    for (int lvl = 0; lvl < NUM_LEVEL; ++lvl) {
        const float ox  = op[lvl * 4 + 0], oy  = op[lvl * 4 + 1];
        const float osw = op[lvl * 4 + 2], osh = op[lvl * 4 + 3];
        const float bcx = cx + ox  * 0.125f * sw;
        const float bcy = cy + oy  * 0.125f * sh;
        const float bw  = fmaxf(sw + osw * 0.125f * sw, 0.f);
        const float bh  = fmaxf(sh + osh * 0.125f * sh, 0.f);
        const float vrx = vratio[b * 8 + lvl * 2 + 0];
        const float vry = vratio[b * 8 + lvl * 2 + 1];
        const int W = HW[lvl], H = HW[lvl];
        const float* vl = vproj + ((size_t)(b * NUM_HEAD + head) * L_LEN + ST[lvl]) * HEAD_DIM;
#pragma unroll
        for (int p = 0; p < NUM_POINT; ++p) {
            const float gx = (bcx + kidx[p * 2 + 0] * bw) * vrx;
            const float gy = (bcy + kidx[p * 2 + 1] * bh) * vry;
            const float x = gx * (float)W - 0.5f;
            const float y = gy * (float)H - 0.5f;
            const float x0f = floorf(x), y0f = floorf(y);
            const float dx = x - x0f, dy = y - y0f;
            const int x0 = (int)x0f, y0 = (int)y0f;
            const float a = ap[lvl * 4 + p];
            const float w00 = (1.f - dx) * (1.f - dy);
            const float w10 = dx * (1.f - dy);
            const float w01 = (1.f - dx) * dy;
            const float w11 = dx * dy;
            float s = 0.f;
            const bool xv0 = (x0 >= 0) && (x0 < W);
            const bool xv1 = (x0 + 1 >= 0) && (x0 + 1 < W);
            const bool yv0 = (y0 >= 0) && (y0 < H);
            const bool yv1 = (y0 + 1 >= 0) && (y0 + 1 < H);
            if (xv0 && yv0) s += w00 * vl[(size_t)(y0 * W + x0) * HEAD_DIM + lane];
            if (xv1 && yv0) s += w10 * vl[(size_t)(y0 * W + x0 + 1) * HEAD_DIM + lane];
            if (xv0 && yv1) s += w01 * vl[(size_t)((y0 + 1) * W + x0) * HEAD_DIM + lane];
            if (xv1 && yv1) s += w11 * vl[(size_t)((y0 + 1) * W + x0 + 1) * HEAD_DIM + lane];
            acc += a * s;
        }
    }
    out_pre[(size_t)q * D_MODEL + head * HEAD_DIM + lane] = acc;
}

// ----------------------------------------------------------------------
extern "C" void kernel_launch(void* const* d_in, const int* in_sizes, int n_in,
                              void* d_out, int out_size, void* d_ws, size_t ws_size,
                              hipStream_t stream)
{
    const float*   query  = (const float*)d_in[0];
    const float*   value  = (const float*)d_in[1];
    // d_in[2] v_shape, d_in[3] v_start_index: compile-time constants (hardcoded)
    const uint8_t* vmask  = (const uint8_t*)d_in[4];   // all-zero bytes either way
    const float*   vratio = (const float*)d_in[5];
    const float*   refw   = (const float*)d_in[6];
    const float*   Wv     = (const float*)d_in[7];
    const float*   bv     = (const float*)d_in[8];
    const float*   Wo     = (const float*)d_in[9];
    const float*   bo     = (const float*)d_in[10];
    const float*   Wbox   = (const float*)d_in[11];
    const float*   bbox   = (const float*)d_in[12];
    const float*   Wattn  = (const float*)d_in[13];
    const float*   battn  = (const float*)d_in[14];
    const float*   kidx   = (const float*)d_in[15];

    float* out      = (float*)d_out;                         // (B, L1, 256)
    float* attn_out = out + (size_t)M_ROWS * D_MODEL;        // (B, L1, 8, 4, 2, 2)

    float* vproj  = (float*)d_ws;                            // (B, head, L2, 32)
    float* offbuf = vproj  + (size_t)M_ROWS * D_MODEL;       // (M, 128)
    float* outpre = offbuf + (size_t)M_ROWS * 128;           // (M, 256)

    const dim3 wave(32, 1, 1);
    const int mt = M_ROWS / 16;                              // 5440 row tiles

    k_vproj    <<<dim3(mt, D_MODEL / (16 * NT)), wave, 0, stream>>>(value, Wv, bv, vmask, vproj);
    k_attn     <<<dim3(mt, 128     / (16 * NT)), wave, 0, stream>>>(query, Wattn, battn, attn_out);
    k_gemm_bias<<<dim3(mt, 128     / (16 * NT)), wave, 0, stream>>>(query, Wbox, bbox, offbuf, 128);
    k_sample   <<<M_ROWS, NUM_HEAD * 32, 0, stream>>>(vproj, attn_out, offbuf,
                                                      refw, vratio, kidx, outpre);
    k_gemm_bias<<<dim3(mt, D_MODEL / (16 * NT)), wave, 0, stream>>>(outpre, Wo, bo, out, D_MODEL);
}